// HotAnchorLayer_30339648978946
// MI455X (gfx1250) — compile-verified
//
#include <hip/hip_runtime.h>
#include <cstdint>
#include <cstddef>

#define B_      8
#define C_      256
#define HW_     40000   // 200*200
#define W_      200
#define COUNTS_ 1000
#define NANCH   12

typedef __attribute__((ext_vector_type(2))) float v2f;
typedef __attribute__((ext_vector_type(8))) float v8f;

// ---------------- Kernel 1: per-(b,c) mean over H*W ----------------
__global__ void mean_kernel(const float* __restrict__ x, float* __restrict__ mean) {
  __shared__ float red[256];
  const int blk = blockIdx.x;                 // blk = b*256 + c
  const float* p = x + (size_t)blk * HW_;
  float s = 0.0f;
  for (int i = threadIdx.x; i < HW_; i += 256) s += p[i];   // coalesced
  red[threadIdx.x] = s;
  __syncthreads();
  for (int off = 128; off > 0; off >>= 1) {
    if ((int)threadIdx.x < off) red[threadIdx.x] += red[threadIdx.x + off];
    __syncthreads();
  }
  if (threadIdx.x == 0) mean[blk] = red[0] / (float)HW_;
}

// ---------------- Kernel 2: heat via V_WMMA_F32_16X16X4_F32 ----------------
// heat[b][p] = sum_c |x[b][c][p] - mean[b][c]|
// One wave owns a 16-pixel tile. A(16x4) = |x-mean| for 4 channels,
// B(4x16) = ones, so D accumulates the channel sums (replicated per column).
// ISA A-layout (32-bit 16x4): lanes 0-15 hold (M=lane, K=0|1) in v0|v1,
// lanes 16-31 hold (M=lane-16, K=2|3).
__global__ void heat_kernel(const float* __restrict__ x,
                            const float* __restrict__ mean,
                            float* __restrict__ heat) {
  __shared__ float smean[C_];
  const int b = blockIdx.y;
  const int tid = threadIdx.x;
  smean[tid] = mean[b * C_ + tid];            // blockDim.x == 256 == C_
  __syncthreads();

  const int wave = tid >> 5;
  const int lane = tid & 31;
  const int tile = blockIdx.x * 8 + wave;     // uniform within wave
  if (tile >= HW_ / 16) return;               // EXEC stays all-ones per wave
  const int p0 = tile * 16;
  const int m  = lane & 15;                   // pixel row of A
  const int k2 = (lane >> 4) * 2;             // channel pair base: 0 or 2

  const float* xb = x + (size_t)b * C_ * HW_;
  v8f acc = {};
  v2f bones; bones[0] = 1.0f; bones[1] = 1.0f;

  for (int c = 0; c < C_; c += 4) {
    const int c0 = c + k2;
    v2f a;
    a[0] = fabsf(xb[(size_t)c0 * HW_ + p0 + m] - smean[c0]);
    a[1] = fabsf(xb[(size_t)(c0 + 1) * HW_ + p0 + m] - smean[c0 + 1]);
    acc = __builtin_amdgcn_wmma_f32_16x16x4_f32(false, a, false, bones,
                                                (short)0, acc, false, false);
  }

  // D layout: VGPR v, lanes 0-15 -> (M=v, N=lane); lanes 16-31 -> (M=v+8).
  // Every column holds the row sum, so lane 0 extracts pixels 0..7 and
  // lane 16 extracts pixels 8..15.
  float* hb = heat + (size_t)b * HW_ + p0;
  if (lane == 0) {
#pragma unroll
    for (int i = 0; i < 8; ++i) hb[i] = acc[i];
  } else if (lane == 16) {
#pragma unroll
    for (int i = 0; i < 8; ++i) hb[8 + i] = acc[i];
  }
}

// ---------------- Kernel 3: radix-select kth, pick indices, emit boxes ----
__global__ __launch_bounds__(1024)
void select_kernel(const float* __restrict__ heat, float* __restrict__ out) {
  __shared__ int   hist[256];
  __shared__ int   wsum[32];
  __shared__ int   sel[COUNTS_];
  __shared__ int   sstate[2];               // {bin, rem}
  __shared__ float hh[NANCH], hwd[NANCH];   // half-height / half-width

  const int b    = blockIdx.x;
  const int tid  = threadIdx.x;
  const int lane = tid & 31;
  const int wid  = tid >> 5;
  const float* hb = heat + (size_t)b * HW_;

  if (tid < NANCH) {
    // meshgrid(SCALES, RATIOS).flatten(): a -> (ratio = a/4, scale = a%4)
    const float SC[4] = {32.0f, 64.0f, 128.0f, 256.0f};
    const float SQ[3] = {0.70710678118654752440f, 1.0f, 1.41421356237309504880f};
    const float s = SC[tid & 3];
    const float q = SQ[tid >> 2];
    hh[tid]  = 0.5f * (s / q);              // 0.5 * height = 0.5 * s/sqrt(r)
    hwd[tid] = 0.5f * (s * q);              // 0.5 * width  = 0.5 * s*sqrt(r)
  }

  // ---- 4-pass radix select of the COUNTS-th largest value ----
  // heat >= 0 so float bits order exactly as uint32.
  unsigned prefix = 0;
  int rem = COUNTS_;
  for (int pass = 0; pass < 4; ++pass) {
    const int shift = 24 - 8 * pass;
    if (tid < 256) hist[tid] = 0;
    __syncthreads();
    for (int p = tid; p < HW_; p += 1024) {
      const unsigned bits = __float_as_uint(hb[p]);
      const bool match = (pass == 0) || ((bits >> (shift + 8)) == prefix);
      if (match) atomicAdd(&hist[(bits >> shift) & 255u], 1);
    }
    __syncthreads();
    if (tid == 0) {
      int cum = 0, bin = 0;
      for (int bb = 255; bb >= 0; --bb) {
        const int c = hist[bb];
        if (cum + c >= rem) { bin = bb; break; }
        cum += c;
      }
      sstate[0] = bin;
      sstate[1] = rem - cum;
    }
    __syncthreads();
    prefix = (prefix << 8) | (unsigned)sstate[0];
    rem = sstate[1];
    __syncthreads();
  }
  const float kth = __uint_as_float(prefix);

  // ---- first COUNTS indices (ascending) with heat >= kth ----
  // Matches reference: score = -idx on eligible, top_k -> ascending indices.
  int base = 0;
  for (int chunk = 0; chunk < HW_; chunk += 1024) {
    const int p = chunk + tid;
    const bool elig = (p < HW_) && (hb[p] >= kth);
    const unsigned bal = __builtin_amdgcn_ballot_w32(elig);
    if (lane == 0) wsum[wid] = __popc(bal);
    __syncthreads();
    int pre = 0, tot = 0;
    for (int w = 0; w < 32; ++w) {
      const int c = wsum[w];
      tot += c;
      if (w < wid) pre += c;
    }
    const int rank = base + pre + __popc(bal & ((1u << lane) - 1u));
    if (elig && rank < COUNTS_) sel[rank] = p;
    base += tot;                              // uniform across block
    __syncthreads();
    if (base >= COUNTS_) break;               // uniform break
  }
  __syncthreads();

  // ---- emit boxes: out[b][cnt*12 + a][0..3] ----
  float* ob = out + (size_t)b * COUNTS_ * NANCH * 4;
  for (int i = tid; i < COUNTS_ * NANCH; i += 1024) {
    const int cnt = i / NANCH;
    const int a   = i - cnt * NANCH;
    const int pix = sel[cnt];
    const float y = (float)(pix / W_) * 4.0f;   // stride_y = 800/200
    const float x = (float)(pix % W_) * 4.0f;   // stride_x = 800/200
    float v0 = (y - hh[a])  / 800.0f;
    float v1 = (x - hwd[a]) / 800.0f;
    float v2 = (y + hh[a])  / 800.0f;
    float v3 = (x + hwd[a]) / 800.0f;
    v0 = fminf(fmaxf(v0, 0.0f), 1.0f);
    v1 = fminf(fmaxf(v1, 0.0f), 1.0f);
    v2 = fminf(fmaxf(v2, 0.0f), 1.0f);
    v3 = fminf(fmaxf(v3, 0.0f), 1.0f);
    ((float4*)ob)[i] = make_float4(v0, v1, v2, v3);
  }
}

extern "C" void kernel_launch(void* const* d_in, const int* in_sizes, int n_in,
                              void* d_out, int out_size, void* d_ws, size_t ws_size,
                              hipStream_t stream) {
  (void)in_sizes; (void)n_in; (void)out_size; (void)ws_size;
  const float* x0 = (const float*)d_in[0];
  float* out  = (float*)d_out;
  float* mean = (float*)d_ws;                 // B_*C_ floats (8 KB)
  float* heat = mean + B_ * C_;               // B_*HW_ floats (1.25 MB)

  mean_kernel<<<dim3(B_ * C_), dim3(256), 0, stream>>>(x0, mean);
  heat_kernel<<<dim3((HW_ / 16 + 7) / 8, B_), dim3(256), 0, stream>>>(x0, mean, heat);
  select_kernel<<<dim3(B_), dim3(1024), 0, stream>>>(heat, out);
}